// DSFM_83451214561984
// MI455X (gfx1250) — compile-verified
//
#include <hip/hip_runtime.h>
#include <math.h>

typedef __attribute__((ext_vector_type(2))) float v2f;
typedef __attribute__((ext_vector_type(8))) float v8f;

#define Bn 32
#define Hn 224
#define Wn 224
#define Pn (Hn * Wn)          // 50176
#define Gn 64
#define En 128
#define NCn 10
#define KPAD 128              // 121 gabor taps padded to 128
#define INV_BN 0.99999500003749981f   // 1/sqrt(1+1e-5)

// ---------------- Stage 1: 1x1 convs, BN, sigmoid ----------------
__global__ void k_prep(const float* __restrict__ x,
                       const float* __restrict__ gray_w,
                       const float* __restrict__ bott_w,
                       const float* __restrict__ bott_b,
                       const float* __restrict__ bn2_g,
                       const float* __restrict__ bn2_b,
                       float* __restrict__ gray,
                       float* __restrict__ ybuf) {
  int idx = blockIdx.x * blockDim.x + threadIdx.x;
  if (idx >= Bn * Pn) return;
  int b = idx / Pn, p = idx % Pn;
  const float* xb = x + (size_t)b * 3 * Pn + p;
  float x0 = xb[0], x1 = xb[Pn], x2 = xb[2 * Pn];
  gray[idx] = x0 * gray_w[0] + x1 * gray_w[1] + x2 * gray_w[2];
  float v = x0 * bott_w[0] + x1 * bott_w[1] + x2 * bott_w[2] + bott_b[0];
  v = v * INV_BN * bn2_g[0] + bn2_b[0];
  ybuf[idx] = 1.0f / (1.0f + expf(-v));
}

// ---------------- Stage 2: eroded = exp(box3x3(log(clip(y)))) ----------------
__global__ void k_erode(const float* __restrict__ ybuf, float* __restrict__ erod) {
  int idx = blockIdx.x * blockDim.x + threadIdx.x;
  if (idx >= Bn * Pn) return;
  int b = idx / Pn, p = idx % Pn;
  int h = p / Wn, w = p % Wn;
  const float* yb = ybuf + (size_t)b * Pn;
  float s = 0.0f;
  for (int dh = -1; dh <= 1; ++dh)
    for (int dw = -1; dw <= 1; ++dw) {
      int hh = h + dh, ww = w + dw;
      if (hh >= 0 && hh < Hn && ww >= 0 && ww < Wn)
        s += logf(fmaxf(yb[hh * Wn + ww], 1e-6f));
    }
  erod[idx] = expf(s);
}

// ---------------- Stage 3: opened, skeleton = relu(y - opened), in-place into ybuf ----------------
__global__ void k_open_skel(const float* __restrict__ erod, float* __restrict__ ybuf) {
  int idx = blockIdx.x * blockDim.x + threadIdx.x;
  if (idx >= Bn * Pn) return;
  int b = idx / Pn, p = idx % Pn;
  int h = p / Wn, w = p % Wn;
  const float* eb = erod + (size_t)b * Pn;
  float s = 0.0f;
  for (int dh = -1; dh <= 1; ++dh)
    for (int dw = -1; dw <= 1; ++dw) {
      int hh = h + dh, ww = w + dw;
      if (hh >= 0 && hh < Hn && ww >= 0 && ww < Wn)
        s += logf(fmaxf(1.0f - eb[hh * Wn + ww], 1e-6f));
    }
  float opened = 1.0f - expf(s);
  ybuf[idx] = fmaxf(ybuf[idx] - opened, 0.0f);
}

// ---------------- Stage 4: per-batch max & sum(exp) ----------------
__global__ void k_stats(const float* __restrict__ skel, float* __restrict__ stats) {
  __shared__ float red[256];
  int b = blockIdx.x, tid = threadIdx.x;
  const float* sb = skel + (size_t)b * Pn;
  float m = -1e30f;
  for (int p = tid; p < Pn; p += 256) m = fmaxf(m, sb[p]);
  red[tid] = m; __syncthreads();
  for (int s = 128; s > 0; s >>= 1) { if (tid < s) red[tid] = fmaxf(red[tid], red[tid + s]); __syncthreads(); }
  float mv = red[0]; __syncthreads();
  float acc = 0.0f;
  for (int p = tid; p < Pn; p += 256) acc += expf(sb[p] - mv);
  red[tid] = acc; __syncthreads();
  for (int s = 128; s > 0; s >>= 1) { if (tid < s) red[tid] += red[tid + s]; __syncthreads(); }
  if (tid == 0) { stats[2 * b] = mv; stats[2 * b + 1] = red[0]; }
}

// ---------------- Stage 5: softmax weights, in-place ----------------
__global__ void k_wspatial(float* __restrict__ skel, const float* __restrict__ stats) {
  int idx = blockIdx.x * blockDim.x + threadIdx.x;
  if (idx >= Bn * Pn) return;
  int b = idx / Pn;
  skel[idx] = expf(skel[idx] - stats[2 * b]) / stats[2 * b + 1];
}

// ---------------- Stage 6: Gabor kernel table ker[64][128] (k>=121 zero) ----------------
__global__ void k_gabor(const float* __restrict__ theta, const float* __restrict__ sigma,
                        const float* __restrict__ lam, const float* __restrict__ psi,
                        const float* __restrict__ gamma, float* __restrict__ ker) {
  int idx = blockIdx.x * blockDim.x + threadIdx.x;
  if (idx >= Gn * KPAD) return;
  int g = idx / KPAD, k = idx % KPAD;
  if (k >= 121) { ker[idx] = 0.0f; return; }
  int i = k / 11, j = k % 11;
  float yg = (float)(i - 5), xg = (float)(j - 5);
  float t = theta[g], s = sigma[g], L = lam[g], p = psi[g], gm = gamma[g];
  float ct = cosf(t), st = sinf(t);
  float xp = xg * ct + yg * st;
  float yp = -xg * st + yg * ct;
  float env = expf(-(xp * xp + gm * gm * yp * yp) / (2.0f * s * s + 1e-6f));
  float car = cosf(2.0f * 3.14159265358979323846f * xp / (L + 1e-6f) + p);
  ker[idx] = env * car;
}

// ---------------- Stage 7: S[b][k] = sum_p w[b,p] * gray[b, p shifted by tap k] ----------------
__global__ void k_S(const float* __restrict__ gray, const float* __restrict__ wsp,
                    float* __restrict__ S) {
  __shared__ float red[256];
  int id = blockIdx.x;          // b*128 + k
  int b = id / KPAD, k = id % KPAD;
  int tid = threadIdx.x;
  if (k >= 121) { if (tid == 0) S[id] = 0.0f; return; }
  int ky = k / 11 - 5, kx = k % 11 - 5;
  const float* gb = gray + (size_t)b * Pn;
  const float* wb = wsp + (size_t)b * Pn;
  float acc = 0.0f;
  for (int p = tid; p < Pn; p += 256) {
    int h = p / Wn, w = p % Wn;
    int gh = h + ky, gw = w + kx;
    if (gh >= 0 && gh < Hn && gw >= 0 && gw < Wn)
      acc += wb[p] * gb[gh * Wn + gw];
  }
  red[tid] = acc; __syncthreads();
  for (int s = 128; s > 0; s >>= 1) { if (tid < s) red[tid] += red[tid + s]; __syncthreads(); }
  if (tid == 0) S[id] = red[0];
}

// ---------------- Stage 8: WMMA head: pooled = S*ker^T; emb = relu(BN(pooled*Ew)); logits ----------------
// One workgroup, 8 waves (wave32). Uses V_WMMA_F32_16X16X4_F32 (exact fp32 MAC).
// A 16x4 layout: lanes 0-15 row M=lane {K=k0,k0+1}; lanes 16-31 row M=lane-16 {K=k0+2,k0+3}.
// B 4x16 layout (symmetric): lanes 0-15 col N=lane rows {k0,k0+1}; lanes 16-31 col N=lane-16 rows {k0+2,k0+3}.
// C/D 16x16: VGPR i -> (M=i, N=lane) lanes 0-15; (M=i+8, N=lane-16) lanes 16-31.
__global__ void k_head(const float* __restrict__ ker, const float* __restrict__ S,
                       const float* __restrict__ emb_w, const float* __restrict__ emb_b,
                       const float* __restrict__ bn1_g, const float* __restrict__ bn1_b,
                       const float* __restrict__ cls_w, const float* __restrict__ cls_b,
                       float* __restrict__ out) {
  __shared__ float pooled[Bn * Gn];   // 32x64
  __shared__ float emb[Bn * En];      // 32x128
  int tid = threadIdx.x;
  int lane = tid & 31, wv = tid >> 5;
  int l16 = lane & 15;
  int koff = (lane >= 16) ? 2 : 0;
  int moff = (lane >= 16) ? 8 : 0;

  // ---- GEMM1: pooled(32x64) = S(32x128) @ ker^T(128x64); wave wv -> tile (mt=wv/4, nt=wv%4)
  {
    int mt = wv >> 2, nt = wv & 3;
    int row = mt * 16 + l16;
    int col = nt * 16 + l16;     // output col = gabor filter g
    v8f c = {0.f, 0.f, 0.f, 0.f, 0.f, 0.f, 0.f, 0.f};
    for (int kk = 0; kk < KPAD / 4; ++kk) {
      int k0 = kk * 4 + koff;
      v2f a, bm;
      a.x = S[row * KPAD + k0];       a.y = S[row * KPAD + k0 + 1];
      bm.x = ker[col * KPAD + k0];    bm.y = ker[col * KPAD + k0 + 1];  // B[k][g]=ker[g][k]
      c = __builtin_amdgcn_wmma_f32_16x16x4_f32(false, a, false, bm, (short)0, c, false, false);
    }
    for (int i = 0; i < 8; ++i)
      pooled[(mt * 16 + i + moff) * Gn + col] = c[i];
  }
  __syncthreads();

  // ---- GEMM2: emb(32x128) = pooled(32x64) @ emb_w(64x128), + bias/BN/relu epilogue
  for (int t = wv; t < 16; t += 8) {
    int mt = t >> 3, nt = t & 7;
    int row = mt * 16 + l16;
    int colE = nt * 16 + l16;
    v8f c = {0.f, 0.f, 0.f, 0.f, 0.f, 0.f, 0.f, 0.f};
    for (int kk = 0; kk < Gn / 4; ++kk) {
      int k0 = kk * 4 + koff;
      v2f a, bm;
      a.x = pooled[row * Gn + k0];      a.y = pooled[row * Gn + k0 + 1];
      bm.x = emb_w[k0 * En + colE];     bm.y = emb_w[(k0 + 1) * En + colE];
      c = __builtin_amdgcn_wmma_f32_16x16x4_f32(false, a, false, bm, (short)0, c, false, false);
    }
    float sc = INV_BN * bn1_g[colE];
    float sh = bn1_b[colE];
    float eb = emb_b[colE];
    for (int i = 0; i < 8; ++i) {
      float v = (c[i] + eb) * sc + sh;
      emb[(mt * 16 + i + moff) * En + colE] = fmaxf(v, 0.0f);
    }
  }
  __syncthreads();

  // ---- GEMM3: logits(32x10, N padded to 16) = emb(32x128) @ cls_w(128x10); waves 0,1
  if (wv < 2) {
    int mt = wv;
    int row = mt * 16 + l16;
    int col = l16;
    v8f c = {0.f, 0.f, 0.f, 0.f, 0.f, 0.f, 0.f, 0.f};
    for (int kk = 0; kk < En / 4; ++kk) {
      int k0 = kk * 4 + koff;
      v2f a, bm;
      a.x = emb[row * En + k0];  a.y = emb[row * En + k0 + 1];
      bm.x = (col < NCn) ? cls_w[k0 * NCn + col] : 0.0f;
      bm.y = (col < NCn) ? cls_w[(k0 + 1) * NCn + col] : 0.0f;
      c = __builtin_amdgcn_wmma_f32_16x16x4_f32(false, a, false, bm, (short)0, c, false, false);
    }
    if (col < NCn) {
      float cb = cls_b[col];
      for (int i = 0; i < 8; ++i)
        out[(mt * 16 + i + moff) * NCn + col] = c[i] + cb;
    }
  }
}

extern "C" void kernel_launch(void* const* d_in, const int* in_sizes, int n_in,
                              void* d_out, int out_size, void* d_ws, size_t ws_size,
                              hipStream_t stream) {
  const float* x      = (const float*)d_in[0];
  const float* gray_w = (const float*)d_in[1];
  const float* theta  = (const float*)d_in[2];
  const float* sigma  = (const float*)d_in[3];
  const float* lam    = (const float*)d_in[4];
  const float* psi    = (const float*)d_in[5];
  const float* gamma  = (const float*)d_in[6];
  const float* bott_w = (const float*)d_in[7];
  const float* bott_b = (const float*)d_in[8];
  const float* bn2_g  = (const float*)d_in[9];
  const float* bn2_b  = (const float*)d_in[10];
  const float* emb_w  = (const float*)d_in[11];
  const float* emb_b  = (const float*)d_in[12];
  const float* bn1_g  = (const float*)d_in[13];
  const float* bn1_b  = (const float*)d_in[14];
  const float* cls_w  = (const float*)d_in[15];
  const float* cls_b  = (const float*)d_in[16];
  float* out = (float*)d_out;

  char* ws = (char*)d_ws;
  size_t bigN = (size_t)Bn * Pn * sizeof(float);   // 6.42 MB
  float* gray  = (float*)(ws);
  float* ybuf  = (float*)(ws + bigN);              // y -> skeleton -> softmax weights (aliased)
  float* erod  = (float*)(ws + 2 * bigN);
  float* ker   = (float*)(ws + 3 * bigN);                                   // 64*128
  float* Smat  = (float*)(ws + 3 * bigN + Gn * KPAD * sizeof(float));       // 32*128
  float* stats = (float*)(ws + 3 * bigN + (Gn + Bn) * KPAD * sizeof(float)); // 32*2

  int npx = Bn * Pn;
  int nblk = (npx + 255) / 256;

  k_prep<<<nblk, 256, 0, stream>>>(x, gray_w, bott_w, bott_b, bn2_g, bn2_b, gray, ybuf);
  k_erode<<<nblk, 256, 0, stream>>>(ybuf, erod);
  k_open_skel<<<nblk, 256, 0, stream>>>(erod, ybuf);
  k_stats<<<Bn, 256, 0, stream>>>(ybuf, stats);
  k_wspatial<<<nblk, 256, 0, stream>>>(ybuf, stats);
  k_gabor<<<(Gn * KPAD + 255) / 256, 256, 0, stream>>>(theta, sigma, lam, psi, gamma, ker);
  k_S<<<Bn * KPAD, 256, 0, stream>>>(gray, ybuf, Smat);
  k_head<<<1, 256, 0, stream>>>(ker, Smat, emb_w, emb_b, bn1_g, bn1_b, cls_w, cls_b, out);
}